// KirchhoffNet_31052613550405
// MI455X (gfx1250) — compile-verified
//
#include <hip/hip_runtime.h>

// KirchhoffNet edge scatter-add for MI455X (gfx1250, wave32).
// out[n] = (sum_{e: dst=n} cur_e - sum_{e: src=n} cur_e) / THETA
// cur_e  = g_e * relu(t1_e * (v[src_e] - v[dst_e]) + t2_e)
//
// Bandwidth-bound: ~20B/edge streamed, accumulator (400KB) + v (400KB) are
// L2-resident. Use native fire-and-forget GLOBAL_ATOMIC_ADD_F32 and NT loads.

#define KN_THETA 1.0f

typedef float v4f __attribute__((ext_vector_type(4)));
typedef int   v4i __attribute__((ext_vector_type(4)));

// Prefer the raw gfx builtin so we get the non-returning hardware f32 atomic
// (STOREcnt path), not a compare-swap loop.
#if defined(__has_builtin)
#if __has_builtin(__builtin_amdgcn_global_atomic_fadd_f32)
#define KN_HAVE_FADD_BUILTIN 1
#endif
#endif

__device__ __forceinline__ void kn_fadd(float* p, float x) {
#ifdef KN_HAVE_FADD_BUILTIN
  typedef __attribute__((address_space(1))) float gfloat;
  (void)__builtin_amdgcn_global_atomic_fadd_f32((gfloat*)p, x);
#else
  (void)__hip_atomic_fetch_add(p, x, __ATOMIC_RELAXED, __HIP_MEMORY_SCOPE_AGENT);
#endif
}

__global__ __launch_bounds__(256) void kn_zero(float* __restrict__ out, int n) {
  int i = blockIdx.x * 256 + threadIdx.x;
  if (i < n) out[i] = 0.0f;
}

__device__ __forceinline__ void kn_edge(const float* __restrict__ v,
                                        float* __restrict__ out,
                                        int s, int d,
                                        float t1, float t2, float g) {
  // v and out are small (400KB each) -> L2-resident gathers/scatters.
  float diff = v[s] - v[d];
  float cur  = g * fmaxf(fmaf(t1, diff, t2), 0.0f);
  kn_fadd(&out[d],  cur);   // incoming
  kn_fadd(&out[s], -cur);   // outgoing
}

// 4 edges per lane: 128-bit nontemporal loads of each edge stream.
__global__ __launch_bounds__(256) void kn_edges4(
    const float* __restrict__ v,
    const v4i* __restrict__ src4,
    const v4i* __restrict__ dst4,
    const v4f* __restrict__ t14,
    const v4f* __restrict__ t24,
    const v4f* __restrict__ g4,
    float* __restrict__ out,
    int nvec) {
  int i = blockIdx.x * 256 + threadIdx.x;
  if (i >= nvec) return;
  // Single-use streams: NT hint keeps v/accumulator hot in near caches.
  v4i s = __builtin_nontemporal_load(&src4[i]);
  v4i d = __builtin_nontemporal_load(&dst4[i]);
  v4f a = __builtin_nontemporal_load(&t14[i]);
  v4f b = __builtin_nontemporal_load(&t24[i]);
  v4f c = __builtin_nontemporal_load(&g4[i]);
  kn_edge(v, out, s.x, d.x, a.x, b.x, c.x);
  kn_edge(v, out, s.y, d.y, a.y, b.y, c.y);
  kn_edge(v, out, s.z, d.z, a.z, b.z, c.z);
  kn_edge(v, out, s.w, d.w, a.w, b.w, c.w);
}

// Scalar tail for E % 4 edges (E = 3.2M -> empty, kept for generality).
__global__ __launch_bounds__(256) void kn_edges_tail(
    const float* __restrict__ v,
    const int* __restrict__ src,
    const int* __restrict__ dst,
    const float* __restrict__ t1,
    const float* __restrict__ t2,
    const float* __restrict__ g,
    float* __restrict__ out,
    int start, int n_edges) {
  int i = start + blockIdx.x * 256 + threadIdx.x;
  if (i < n_edges) kn_edge(v, out, src[i], dst[i], t1[i], t2[i], g[i]);
}

__global__ __launch_bounds__(256) void kn_scale(float* __restrict__ out, int n,
                                                float s) {
  int i = blockIdx.x * 256 + threadIdx.x;
  if (i < n) out[i] *= s;
}

extern "C" void kernel_launch(void* const* d_in, const int* in_sizes, int n_in,
                              void* d_out, int out_size, void* d_ws, size_t ws_size,
                              hipStream_t stream) {
  // setup_inputs order: t, v, src, dst, theta_sd_1, theta_sd_2, conductance
  const float* v   = (const float*)d_in[1];
  const int*   src = (const int*)  d_in[2];
  const int*   dst = (const int*)  d_in[3];
  const float* t1  = (const float*)d_in[4];
  const float* t2  = (const float*)d_in[5];
  const float* g   = (const float*)d_in[6];
  float* out = (float*)d_out;

  const int n_nodes = out_size;
  const int n_edges = in_sizes[2];

  // 1) Zero the accumulator (d_out is poisoned by the harness).
  kn_zero<<<(n_nodes + 255) / 256, 256, 0, stream>>>(out, n_nodes);

  // 2) Signed scatter-add over all edges (4 edges / lane, b128 NT loads).
  const int nvec = n_edges >> 2;
  if (nvec > 0) {
    kn_edges4<<<(nvec + 255) / 256, 256, 0, stream>>>(
        v, (const v4i*)src, (const v4i*)dst,
        (const v4f*)t1, (const v4f*)t2, (const v4f*)g, out, nvec);
  }
  const int rem = n_edges - (nvec << 2);
  if (rem > 0) {
    kn_edges_tail<<<1, 256, 0, stream>>>(v, src, dst, t1, t2, g, out,
                                         nvec << 2, n_edges);
  }

  // 3) Divide by THETA (identity for THETA == 1.0 -> skip the pass).
  if (KN_THETA != 1.0f) {
    kn_scale<<<(n_nodes + 255) / 256, 256, 0, stream>>>(out, n_nodes,
                                                        1.0f / KN_THETA);
  }
}